// NeuralMemory_67216238182780
// MI455X (gfx1250) — compile-verified
//
#include <hip/hip_runtime.h>
#include <cstdint>

// Problem constants (match reference)
constexpr int Bm = 8;       // batch
constexpr int Sm = 1024;    // sequence
constexpr int Dm = 768;     // width
constexpr int NTm = Dm / 16;        // 48 16-wide tiles per dim
#define THETA_C 0.01f

typedef __attribute__((ext_vector_type(2))) float v2f;
typedef __attribute__((ext_vector_type(8))) float v8f;

static __device__ __forceinline__ v8f wmma4(v2f a, v2f b, v8f c) {
  // D = A(16x4,f32) * B(4x16,f32) + C(16x16,f32)
  return __builtin_amdgcn_wmma_f32_16x16x4_f32(false, a, false, b, (short)0, c,
                                               false, false);
}

static __device__ __forceinline__ float sigmoidf_(float x) {
  return 1.f / (1.f + __expf(-x));
}

// ---------------------------------------------------------------------------
// Generic GEMM: Y[M,N] = act( X[M,K] @ W[N,K]^T + bias[N] )
// One 16x16 tile per wave, fp32 WMMA 16x16x4, grid-stride over tiles.
// ---------------------------------------------------------------------------
__global__ __launch_bounds__(256) void gemm_kernel(
    const float* __restrict__ X, const float* __restrict__ W,
    const float* __restrict__ bias, float* __restrict__ Y,
    int M, int N, int K, int act)
{
  const int lane = threadIdx.x & 31;
  const int wave = threadIdx.x >> 5;
  const int li   = lane & 15;          // row/col within tile
  const int kb   = (lane >> 4) << 1;   // k sub-offset 0 or 2 (A/B f32 layout)
  const int hi8  = (lane >> 4) << 3;   // +8 row offset for C layout
  const int wid  = blockIdx.x * (blockDim.x >> 5) + wave;
  const int nw   = gridDim.x * (blockDim.x >> 5);
  const int tm = M >> 4, tn = N >> 4;
  for (int t = wid; t < tm * tn; t += nw) {
    const int ti = t / tn, tj = t % tn;
    const float* Xp = X + (size_t)(ti * 16 + li) * K;
    const float* Wp = W + (size_t)(tj * 16 + li) * K;
    v8f c = {0.f,0.f,0.f,0.f,0.f,0.f,0.f,0.f};
    for (int k = 0; k < K; k += 4) {
      v2f a = *(const v2f*)(Xp + k + kb);
      v2f b = *(const v2f*)(Wp + k + kb);
      c = wmma4(a, b, c);
    }
    const int col = tj * 16 + li;
    const float bs = bias[col];
    for (int i = 0; i < 8; ++i) {
      float v = c[i] + bs;
      if (act) v = v * sigmoidf_(v);   // SiLU
      Y[(size_t)(ti * 16 + i + hi8) * N + col] = v;
    }
  }
}

// ---------------------------------------------------------------------------
// Gate MLP: per row (b,s): sigmoid( silu(x.W1^T + b1) . w2 + b2 )
// GH == 32 == wave32: lane h owns hidden unit h. gtmp[g][b*S+s]
// ---------------------------------------------------------------------------
__global__ __launch_bounds__(256) void gate_kernel(
    const float* __restrict__ x,
    const float* __restrict__ FW1, const float* __restrict__ Fb1,
    const float* __restrict__ Fw2, const float* __restrict__ Fb2,
    const float* __restrict__ DW1, const float* __restrict__ Db1,
    const float* __restrict__ Dw2, const float* __restrict__ Db2,
    float* __restrict__ gtmp)
{
  const int g = blockIdx.y;
  const float* W1 = g ? DW1 : FW1;
  const float* b1 = g ? Db1 : Fb1;
  const float* w2 = g ? Dw2 : Fw2;
  const float* b2 = g ? Db2 : Fb2;
  const int lane = threadIdx.x & 31;
  const int row  = blockIdx.x * 8 + (threadIdx.x >> 5);   // 0 .. B*S-1
  const float* xr = x + (size_t)row * Dm;
  const float* wr = W1 + (size_t)lane * Dm;
  float acc = 0.f;
  for (int d = 0; d < Dm; d += 4) {
    acc += xr[d] * wr[d] + xr[d+1] * wr[d+1]
         + xr[d+2] * wr[d+2] + xr[d+3] * wr[d+3];
  }
  acc += b1[lane];
  float p = (acc * sigmoidf_(acc)) * w2[lane];
  for (int off = 16; off > 0; off >>= 1) p += __shfl_down(p, off);
  if (lane == 0) gtmp[(size_t)g * (Bm * Sm) + row] = sigmoidf_(p + b2[0]);
}

__global__ void gate_reduce(const float* __restrict__ gtmp,
                            float* __restrict__ alpha, float* __restrict__ eta)
{
  const int s = blockIdx.x * blockDim.x + threadIdx.x;
  if (s >= Sm) return;
  float a = 0.f, e = 0.f;
  for (int b = 0; b < Bm; ++b) {
    a += gtmp[(size_t)b * Sm + s];
    e += gtmp[(size_t)(Bm * Sm) + (size_t)b * Sm + s];
  }
  alpha[s] = a * (1.f / Bm);
  eta[s]   = e * (1.f / Bm);
}

// ---------------------------------------------------------------------------
// Device-wide barrier (48 resident blocks). Release/acquire via __threadfence.
// ---------------------------------------------------------------------------
static __device__ __forceinline__ void gbar(int* cnt, int* gen, int nblk,
                                            int* sgen)
{
  __threadfence();
  __syncthreads();
  if (threadIdx.x == 0) {
    const int target = ++(*sgen);
    if (atomicAdd(cnt, 1) == nblk - 1) {
      atomicExch(cnt, 0);
      __threadfence();
      atomicAdd(gen, 1);
    } else {
      while (__hip_atomic_load(gen, __ATOMIC_RELAXED,
                               __HIP_MEMORY_SCOPE_AGENT) < target) {
        __builtin_amdgcn_s_sleep(4);
      }
    }
  }
  __syncthreads();
  __threadfence();
}

// ---------------------------------------------------------------------------
// Persistent scan kernel: 1024 sequential optimizer steps.
// 48 blocks x 256 threads (8 waves). 4 device barriers per step.
//  P1: z1 = kt@W1^T + b1, h1 = silu(z1); stage padded ktpad  (split-K, 8 waves)
//  P2: d  = (2/(B*D)) * (h1@W2^T + b2 - vt)       (rows >= 8 forced 0)
//  P3: dz1 = (d@W2) * silu'(z1);  b2/Sb2 update
//  P4: rank-16 WMMA outer products -> fused S/P update of W1,W2; b1/Sb1 update
// ---------------------------------------------------------------------------
__global__ __launch_bounds__(256) void scan_kernel(
    const float* __restrict__ kproj, const float* __restrict__ vproj,
    const float* __restrict__ alpha, const float* __restrict__ eta,
    float* __restrict__ W1, float* __restrict__ b1,
    float* __restrict__ W2, float* __restrict__ b2,
    float* __restrict__ SW1, float* __restrict__ Sb1,
    float* __restrict__ SW2, float* __restrict__ Sb2,
    float* __restrict__ z1, float* __restrict__ h1,
    float* __restrict__ dbuf, float* __restrict__ dz1,
    float* __restrict__ ktpad, const float* __restrict__ zrow,
    int* bar_cnt, int* bar_gen, int nblk)
{
  __shared__ float red[8][32][9];   // split-K partial C tiles (padded vs banks)
  __shared__ int s_gen;
  const int lane = threadIdx.x & 31;
  const int wave = threadIdx.x >> 5;
  const int li   = lane & 15;
  const int kb   = (lane >> 4) << 1;
  const int hi8  = (lane >> 4) << 3;
  if (threadIdx.x == 0) s_gen = 0;
  __syncthreads();

  const int n0 = blockIdx.x * 16;              // this block's column slice
  const int gw = blockIdx.x * 8 + wave;        // global wave id (0..383)
  const float SC = 2.0f / (float)(Bm * Dm);
  const bool mv = li < Bm;                     // lane maps to a real batch row

  for (int st = 0; st < Sm; ++st) {
    const float a_t = alpha[st];
    const float e_t = eta[st];
    const float one_a = 1.f - a_t;
    const float* kt = kproj + (size_t)st * Dm;   // kt[b][d] = kt[b*S*D + d]
    const float* vt = vproj + (size_t)st * Dm;

    // ---- Phase 1 ----
    {
      v8f c = {0.f,0.f,0.f,0.f,0.f,0.f,0.f,0.f};
      // Padded batch rows read a zeroed dummy row: branch-free inner loop.
      const float* ap = mv ? (kt + (size_t)li * (Sm * Dm)) : zrow;
      const float* bp = W1 + (size_t)(n0 + li) * Dm; // B = W1^T
      const int k0 = wave * (Dm / 8);
      for (int k = k0; k < k0 + Dm / 8; k += 4) {
        v2f a = *(const v2f*)(ap + k + kb);
        v2f b = *(const v2f*)(bp + k + kb);
        c = wmma4(a, b, c);
      }
      for (int i = 0; i < 8; ++i) red[wave][lane][i] = c[i];
      __syncthreads();
      if (threadIdx.x < 32) {
        const int col = n0 + li;
        const float bs = b1[col];
        for (int i = 0; i < 8; ++i) {
          float v = bs;
          for (int w = 0; w < 8; ++w) v += red[w][lane][i];
          const int row = i + hi8;
          z1[row * Dm + col] = v;
          h1[row * Dm + col] = v * sigmoidf_(v);
        }
      } else if (threadIdx.x < 64) {
        // Stage zero-padded kt tile for this block's column slice (phase 4 B).
        const int t2 = threadIdx.x - 32;
        for (int e = t2; e < 256; e += 32) {
          const int bb = e >> 4;             // padded batch row 0..15
          const int cc = n0 + (e & 15);
          ktpad[bb * Dm + cc] =
              (bb < Bm) ? kt[(size_t)bb * (Sm * Dm) + cc] : 0.f;
        }
      }
    }
    gbar(bar_cnt, bar_gen, nblk, &s_gen);

    // ---- Phase 2 ----
    {
      v8f c = {0.f,0.f,0.f,0.f,0.f,0.f,0.f,0.f};
      const float* ap = h1 + (size_t)li * Dm;
      const float* bp = W2 + (size_t)(n0 + li) * Dm; // B = W2^T
      const int k0 = wave * (Dm / 8);
      for (int k = k0; k < k0 + Dm / 8; k += 4) {
        v2f a = *(const v2f*)(ap + k + kb);
        v2f b = *(const v2f*)(bp + k + kb);
        c = wmma4(a, b, c);
      }
      for (int i = 0; i < 8; ++i) red[wave][lane][i] = c[i];
      __syncthreads();
      if (threadIdx.x < 32) {
        const int col = n0 + li;
        const float bs = b2[col];
        for (int i = 0; i < 8; ++i) {
          float v = bs;
          for (int w = 0; w < 8; ++w) v += red[w][lane][i];
          const int row = i + hi8;
          float dv = 0.f;
          if (row < Bm) dv = SC * (v - vt[(size_t)row * (Sm * Dm) + col]);
          dbuf[row * Dm + col] = dv;     // padded rows exactly zero
        }
      }
    }
    gbar(bar_cnt, bar_gen, nblk, &s_gen);

    // ---- Phase 3 ----
    {
      v8f c = {0.f,0.f,0.f,0.f,0.f,0.f,0.f,0.f};
      const float* ap = dbuf + (size_t)li * Dm;
      const int k0 = wave * (Dm / 8);
      for (int k = k0; k < k0 + Dm / 8; k += 4) {
        v2f a = *(const v2f*)(ap + k + kb);
        v2f b;                                   // B = W2 (row-major K x N)
        b.x = W2[(size_t)(k + kb) * Dm + n0 + li];
        b.y = W2[(size_t)(k + kb + 1) * Dm + n0 + li];
        c = wmma4(a, b, c);
      }
      for (int i = 0; i < 8; ++i) red[wave][lane][i] = c[i];
      __syncthreads();
      if (threadIdx.x < 32) {
        const int col = n0 + li;
        for (int i = 0; i < 8; ++i) {
          float dh = 0.f;
          for (int w = 0; w < 8; ++w) dh += red[w][lane][i];
          const int row = i + hi8;
          const float z = z1[row * Dm + col];
          const float sg = sigmoidf_(z);
          dz1[row * Dm + col] = dh * (sg * (1.f + z * (1.f - sg)));
        }
      } else if (threadIdx.x < 48) {
        // bias-2 gradient + update for this block's 16-column slice
        const int col = n0 + (threadIdx.x - 32);
        float g = 0.f;
        for (int bb = 0; bb < Bm; ++bb) g += dbuf[bb * Dm + col];
        const float s = e_t * Sb2[col] - THETA_C * g;
        Sb2[col] = s;
        b2[col] = one_a * b2[col] + s;
      }
    }
    gbar(bar_cnt, bar_gen, nblk, &s_gen);

    // ---- Phase 4: weight updates (rank-16 WMMA outer products, fused) ----
    {
      if (threadIdx.x < 16) {
        const int col = n0 + threadIdx.x;
        float g = 0.f;
        for (int bb = 0; bb < Bm; ++bb) g += dz1[bb * Dm + col];
        const float s = e_t * Sb1[col] - THETA_C * g;
        Sb1[col] = s;
        b1[col] = one_a * b1[col] + s;
      }
      for (int tt = gw; tt < 2 * NTm * NTm; tt += 8 * 48) {
        const int w2sel = (tt < NTm * NTm) ? 1 : 0;
        const int t2 = w2sel ? tt : tt - NTm * NTm;
        const int r0 = (t2 / NTm) * 16;
        const int c0 = (t2 % NTm) * 16;
        const float* A = w2sel ? dbuf : dz1;   // A(m,k) = A[k][r0+m], K=16
        const float* Bp = w2sel ? h1 : ktpad;  // B(k,n) = Bp[k][c0+n]
        v8f c = {0.f,0.f,0.f,0.f,0.f,0.f,0.f,0.f};
        for (int k = 0; k < 16; k += 4) {
          const int kk = k + kb;
          v2f a, b;
          a.x = A[kk * Dm + r0 + li];
          a.y = A[(kk + 1) * Dm + r0 + li];
          b.x = Bp[kk * Dm + c0 + li];
          b.y = Bp[(kk + 1) * Dm + c0 + li];
          c = wmma4(a, b, c);
        }
        float* Pw = w2sel ? W2 : W1;
        float* Sw = w2sel ? SW2 : SW1;
        for (int i = 0; i < 8; ++i) {
          const int idx = (r0 + i + hi8) * Dm + c0 + li;
          const float s = e_t * Sw[idx] - THETA_C * c[i];
          Sw[idx] = s;
          Pw[idx] = one_a * Pw[idx] + s;
        }
      }
    }
    gbar(bar_cnt, bar_gen, nblk, &s_gen);
  }
}

// ---------------------------------------------------------------------------
extern "C" void kernel_launch(void* const* d_in, const int* in_sizes, int n_in,
                              void* d_out, int out_size, void* d_ws,
                              size_t ws_size, hipStream_t stream)
{
  const float* x    = (const float*)d_in[0];
  const float* WKw  = (const float*)d_in[1];
  const float* WKb  = (const float*)d_in[2];
  const float* WVw  = (const float*)d_in[3];
  const float* WVb  = (const float*)d_in[4];
  const float* WQw  = (const float*)d_in[5];
  const float* WQb  = (const float*)d_in[6];
  const float* MW1  = (const float*)d_in[7];
  const float* Mb1  = (const float*)d_in[8];
  const float* MW2  = (const float*)d_in[9];
  const float* Mb2  = (const float*)d_in[10];
  const float* FGW1 = (const float*)d_in[11];
  const float* FGb1 = (const float*)d_in[12];
  const float* FGw2 = (const float*)d_in[13];
  const float* FGb2 = (const float*)d_in[14];
  const float* DGW1 = (const float*)d_in[15];
  const float* DGb1 = (const float*)d_in[16];
  const float* DGw2 = (const float*)d_in[17];
  const float* DGb2 = (const float*)d_in[18];

  // Workspace layout (floats)
  int*   ctrl = (int*)d_ws;                       // [0]=cnt, [1]=gen
  float* f    = (float*)d_ws + 64;
  const size_t NP = (size_t)Bm * Sm * Dm;         // 6,291,456
  float* kproj = f; f += NP;                      // reused as hidden buf later
  float* vproj = f; f += NP;                      // reused as q-proj later
  float* W1 = f;  f += (size_t)Dm * Dm;
  float* b1 = f;  f += Dm;
  float* W2 = f;  f += (size_t)Dm * Dm;
  float* b2 = f;  f += Dm;
  float* SW1 = f; f += (size_t)Dm * Dm;           // SW1..zrow contiguous memset
  float* Sb1 = f; f += Dm;
  float* SW2 = f; f += (size_t)Dm * Dm;
  float* Sb2 = f; f += Dm;
  float* zrow = f; f += Dm;                       // zeroed dummy row (padding)
  float* alphaB = f; f += Sm;
  float* etaB   = f; f += Sm;
  float* gtmp   = f; f += (size_t)2 * Bm * Sm;
  float* z1b    = f; f += (size_t)16 * Dm;
  float* h1b    = f; f += (size_t)16 * Dm;
  float* dbufB  = f; f += (size_t)16 * Dm;
  float* dz1B   = f; f += (size_t)16 * Dm;
  float* ktpadB = f; f += (size_t)16 * Dm;
  (void)ws_size; (void)in_sizes; (void)n_in;

  // Deterministic init every call (harness poisons ws once, never restores)
  hipMemsetAsync(ctrl, 0, 256, stream);
  hipMemcpyAsync(W1, MW1, sizeof(float) * Dm * Dm, hipMemcpyDeviceToDevice, stream);
  hipMemcpyAsync(b1, Mb1, sizeof(float) * Dm,      hipMemcpyDeviceToDevice, stream);
  hipMemcpyAsync(W2, MW2, sizeof(float) * Dm * Dm, hipMemcpyDeviceToDevice, stream);
  hipMemcpyAsync(b2, Mb2, sizeof(float) * Dm,      hipMemcpyDeviceToDevice, stream);
  hipMemsetAsync(SW1, 0, sizeof(float) * (2ull * Dm * Dm + 3ull * Dm), stream);

  const int M = Bm * Sm;                   // 8192 rows
  const int gemmBlocks = (M / 16) * (Dm / 16) / 8;   // 1 tile per wave

  // Projections k, v  (q computed after the scan, into vproj's space)
  gemm_kernel<<<gemmBlocks, 256, 0, stream>>>(x, WKw, WKb, kproj, M, Dm, Dm, 0);
  gemm_kernel<<<gemmBlocks, 256, 0, stream>>>(x, WVw, WVb, vproj, M, Dm, Dm, 0);

  // Gates -> alpha[s], eta[s]
  dim3 ggrid(M / 8, 2);
  gate_kernel<<<ggrid, 256, 0, stream>>>(x, FGW1, FGb1, FGw2, FGb2,
                                         DGW1, DGb1, DGw2, DGb2, gtmp);
  gate_reduce<<<(Sm + 255) / 256, 256, 0, stream>>>(gtmp, alphaB, etaB);

  // Sequential optimizer scan (persistent, device-wide barriers)
  scan_kernel<<<48, 256, 0, stream>>>(kproj, vproj, alphaB, etaB,
                                      W1, b1, W2, b2, SW1, Sb1, SW2, Sb2,
                                      z1b, h1b, dbufB, dz1B, ktpadB, zrow,
                                      ctrl, ctrl + 1, 48);

  // q projection (into vproj space; v is dead after the scan)
  gemm_kernel<<<gemmBlocks, 256, 0, stream>>>(x, WQw, WQb, vproj, M, Dm, Dm, 0);
  // Retrieval MLP with final params: hidden (silu) into kproj space, then out
  gemm_kernel<<<gemmBlocks, 256, 0, stream>>>(vproj, W1, b1, kproj, M, Dm, Dm, 1);
  gemm_kernel<<<gemmBlocks, 256, 0, stream>>>(kproj, W2, b2, (float*)d_out,
                                              M, Dm, Dm, 0);
}